// GATLayer_10282151707708
// MI455X (gfx1250) — compile-verified
//
#include <hip/hip_runtime.h>
#include <hip/hip_bf16.h>

// ---------------------------------------------------------------------------
// GAT layer for MI455X (gfx1250, wave32).
//   h   = feat @ W_fc                      [10000,512]   (WMMA f32 16x16x4)
//   e_ns/e_nd = <h, att>                   [10000,4]
//   e   = LeakyReLU(e_ns[src]+e_nd[dst])   [160000,4]
//   softmax over incoming edges (atomicMax key + atomicAdd denom)
//   agg = scatter-add a * h[src]           [10000,512]   (f32 atomics, L2-resident)
//   out = agg @ W_trans + b                [10000,128]   (WMMA f32 16x16x4)
// ---------------------------------------------------------------------------

#define N_NODES 10000
#define N_EDGES 160000
#define DIMF    128
#define HEADS   4
#define HD      (HEADS * DIMF)   // 512

typedef __attribute__((ext_vector_type(2))) float v2f;
typedef __attribute__((ext_vector_type(8))) float v8f;

// monotone float -> uint key for atomicMax-based segment max (sentinel = 0)
__device__ __forceinline__ unsigned f2key(float f) {
    unsigned b = __float_as_uint(f);
    return (b & 0x80000000u) ? ~b : (b | 0x80000000u);
}
__device__ __forceinline__ float key2f(unsigned k) {
    return __uint_as_float((k & 0x80000000u) ? (k & 0x7FFFFFFFu) : ~k);
}

// ---------------------------------------------------------------------------
// GEMM1: h[10000,512] = feat[10000,128] @ W_fc[128,512]
// grid = 625*4 blocks of 256 (8 waves). Each wave: one 16x16 tile, K=128.
// ---------------------------------------------------------------------------
__global__ __launch_bounds__(256) void k_gemm_fc(const float* __restrict__ A,
                                                 const float* __restrict__ B,
                                                 float* __restrict__ C) {
    const int lane   = threadIdx.x & 31;
    const int wave   = threadIdx.x >> 5;
    const int m_tile = blockIdx.x >> 2;                      // 0..624
    const int n_tile = ((blockIdx.x & 3) << 3) + wave;       // 0..31
    const int row    = m_tile * 16 + (lane & 15);
    const int col    = n_tile * 16 + (lane & 15);
    const int khalf  = (lane >> 4) << 1;                     // 0 or 2

    v8f acc = {};
    const float* ap = A + row * DIMF;
    for (int k0 = 0; k0 < DIMF; k0 += 4) {
        const int kk = k0 + khalf;
        v2f a, b;
        a.x = ap[kk];
        a.y = ap[kk + 1];
        b.x = B[kk * HD + col];
        b.y = B[(kk + 1) * HD + col];
        acc = __builtin_amdgcn_wmma_f32_16x16x4_f32(false, a, false, b,
                                                    (short)0, acc, false, false);
    }
    const int mbase = m_tile * 16 + ((lane >> 4) << 3);
#pragma unroll
    for (int v = 0; v < 8; ++v)
        C[(mbase + v) * HD + col] = acc[v];
}

// ---------------------------------------------------------------------------
// Node attention halves: e_ns[n,h] = <h[n,h,:], att_ns[h,:]>, same for _nd.
// One wave per (node, head); 8 waves / block, grid = 5000.
// ---------------------------------------------------------------------------
__global__ __launch_bounds__(256) void k_scores(const float* __restrict__ h,
                                                const float* __restrict__ ans,
                                                const float* __restrict__ andv,
                                                float* __restrict__ ens,
                                                float* __restrict__ endv) {
    const int lane = threadIdx.x & 31;
    const int wave = threadIdx.x >> 5;
    const int pair = blockIdx.x * 8 + wave;      // 0..39999
    const int n    = pair >> 2;
    const int hd   = pair & 3;

    const float* hp = h + n * HD + hd * DIMF;
    const float* a1 = ans + hd * DIMF;
    const float* a2 = andv + hd * DIMF;

    float s1 = 0.f, s2 = 0.f;
#pragma unroll
    for (int i = 0; i < 4; ++i) {
        const int d = lane + 32 * i;
        const float v = hp[d];
        s1 += v * a1[d];
        s2 += v * a2[d];
    }
#pragma unroll
    for (int off = 16; off >= 1; off >>= 1) {
        s1 += __shfl_xor(s1, off, 32);
        s2 += __shfl_xor(s2, off, 32);
    }
    if (lane == 0) {
        ens[pair]  = s1;
        endv[pair] = s2;
    }
}

// ---------------------------------------------------------------------------
// Edge pass 1: e = LeakyReLU(e_ns[src] + e_nd[dst]); segment max via
// ordered-uint atomicMax. One thread per (edge, head): 640000 threads.
// ---------------------------------------------------------------------------
__global__ __launch_bounds__(256) void k_edge_e(const float* __restrict__ ens,
                                                const float* __restrict__ endv,
                                                const int* __restrict__ src,
                                                const int* __restrict__ dst,
                                                float* __restrict__ e_buf,
                                                unsigned* __restrict__ mkey) {
    const int idx = blockIdx.x * 256 + threadIdx.x;
    const int e   = idx >> 2;
    const int hd  = idx & 3;
    const int s   = src[e];
    const int d   = dst[e];
    float v = ens[s * 4 + hd] + endv[d * 4 + hd];
    v = (v > 0.f) ? v : 0.2f * v;
    e_buf[idx] = v;
    atomicMax(&mkey[d * 4 + hd], f2key(v));
}

// ---------------------------------------------------------------------------
// Edge pass 2: ex = exp(e - m[dst]); denom[dst] += ex.
// ---------------------------------------------------------------------------
__global__ __launch_bounds__(256) void k_edge_ex(const float* __restrict__ e_buf,
                                                 const int* __restrict__ dst,
                                                 const unsigned* __restrict__ mkey,
                                                 float* __restrict__ denom,
                                                 float* __restrict__ ex_buf) {
    const int idx = blockIdx.x * 256 + threadIdx.x;
    const int e   = idx >> 2;
    const int hd  = idx & 3;
    const int d   = dst[e];
    const float ex = expf(e_buf[idx] - key2f(mkey[d * 4 + hd]));
    ex_buf[idx] = ex;
    atomicAdd(&denom[d * 4 + hd], ex);
}

// ---------------------------------------------------------------------------
// Message pass: agg[dst,h,:] += (ex/denom[dst,h]) * h[src,h,:]
// One block (512 threads) per edge; coalesced row gather + row atomics,
// both L2-resident (h and agg are 20.5 MB each << 192 MB L2).
// ---------------------------------------------------------------------------
__global__ __launch_bounds__(512) void k_message(const float* __restrict__ h,
                                                 const float* __restrict__ ex_buf,
                                                 const float* __restrict__ denom,
                                                 const int* __restrict__ src,
                                                 const int* __restrict__ dst,
                                                 float* __restrict__ agg) {
    const int e  = blockIdx.x;
    const int t  = threadIdx.x;      // 0..511 -> (head = t>>7, feat = t&127)
    const int hd = t >> 7;
    const int s  = src[e];
    const int d  = dst[e];
    const float a = ex_buf[e * 4 + hd] / denom[d * 4 + hd];
    atomicAdd(&agg[d * HD + t], a * h[s * HD + t]);
}

// ---------------------------------------------------------------------------
// GEMM2: out[10000,128] = agg[10000,512] @ W_trans[512,128] + b_trans
// grid = 625 blocks of 256 (8 waves = 8 N-tiles). K=512.
// ---------------------------------------------------------------------------
__global__ __launch_bounds__(256) void k_gemm_out(const float* __restrict__ A,
                                                  const float* __restrict__ B,
                                                  const float* __restrict__ bias,
                                                  float* __restrict__ C) {
    const int lane   = threadIdx.x & 31;
    const int wave   = threadIdx.x >> 5;             // n_tile 0..7
    const int m_tile = blockIdx.x;                   // 0..624
    const int row    = m_tile * 16 + (lane & 15);
    const int col    = wave * 16 + (lane & 15);
    const int khalf  = (lane >> 4) << 1;

    v8f acc = {};
    const float* ap = A + row * HD;
    for (int k0 = 0; k0 < HD; k0 += 4) {
        const int kk = k0 + khalf;
        v2f a, b;
        a.x = ap[kk];
        a.y = ap[kk + 1];
        b.x = B[kk * DIMF + col];
        b.y = B[(kk + 1) * DIMF + col];
        acc = __builtin_amdgcn_wmma_f32_16x16x4_f32(false, a, false, b,
                                                    (short)0, acc, false, false);
    }
    const float bv = bias[col];
    const int mbase = m_tile * 16 + ((lane >> 4) << 3);
#pragma unroll
    for (int v = 0; v < 8; ++v)
        C[(mbase + v) * DIMF + col] = acc[v] + bv;
}

// ---------------------------------------------------------------------------
extern "C" void kernel_launch(void* const* d_in, const int* in_sizes, int n_in,
                              void* d_out, int out_size, void* d_ws, size_t ws_size,
                              hipStream_t stream) {
    const float* feat    = (const float*)d_in[0];
    const float* W_fc    = (const float*)d_in[1];
    const float* att_ns  = (const float*)d_in[2];
    const float* att_nd  = (const float*)d_in[3];
    const float* W_trans = (const float*)d_in[4];
    const float* b_trans = (const float*)d_in[5];
    const int*   src     = (const int*)d_in[6];
    const int*   dst     = (const int*)d_in[7];
    float*       out     = (float*)d_out;

    // workspace carve-up (floats)
    float*    ws     = (float*)d_ws;
    float*    h      = ws;                         // 5,120,000
    float*    agg    = h + (size_t)N_NODES * HD;   // 5,120,000
    float*    ens    = agg + (size_t)N_NODES * HD; // 40,000
    float*    endv   = ens + N_NODES * HEADS;      // 40,000
    unsigned* mkey   = (unsigned*)(endv + N_NODES * HEADS); // 40,000
    float*    denom  = (float*)(mkey + N_NODES * HEADS);    // 40,000
    float*    e_buf  = denom + N_NODES * HEADS;    // 640,000
    float*    ex_buf = e_buf + (size_t)N_EDGES * HEADS;     // 640,000

    // zero accumulators each call (graph-capture-safe memsets)
    hipMemsetAsync(agg,   0, (size_t)N_NODES * HD * sizeof(float), stream);
    hipMemsetAsync(mkey,  0, (size_t)N_NODES * HEADS * sizeof(unsigned), stream);
    hipMemsetAsync(denom, 0, (size_t)N_NODES * HEADS * sizeof(float), stream);

    k_gemm_fc<<<625 * 4, 256, 0, stream>>>(feat, W_fc, h);
    k_scores<<<(N_NODES * HEADS) / 8, 256, 0, stream>>>(h, att_ns, att_nd, ens, endv);
    k_edge_e<<<(N_EDGES * HEADS) / 256, 256, 0, stream>>>(ens, endv, src, dst, e_buf, mkey);
    k_edge_ex<<<(N_EDGES * HEADS) / 256, 256, 0, stream>>>(e_buf, dst, mkey, denom, ex_buf);
    k_message<<<N_EDGES, 512, 0, stream>>>(h, ex_buf, denom, src, dst, agg);
    k_gemm_out<<<625, 256, 0, stream>>>(agg, W_trans, b_trans, out);
}